// TypedCoords2Volume_446676599379
// MI455X (gfx1250) — compile-verified
//
#include <hip/hip_runtime.h>

#define BOX 120
#define NTYPES 11
#define WIN 125   // (2*D+1)^3 with D=2

// ---------------------------------------------------------------------------
// gfx1250 async-store feature detection (compile-clean on any toolchain).
// Builtin signature (from hipcc diagnostic):
//   void __builtin_amdgcn_global_store_async_from_lds_b128(
//       as1 v4i* gaddr, as3 v4i* lds, imm int offset, imm int cpol)
// ---------------------------------------------------------------------------
#if defined(__has_builtin)
#if __has_builtin(__builtin_amdgcn_global_store_async_from_lds_b128)
#define USE_ASYNC_STORE 1
#endif
#if __has_builtin(__builtin_amdgcn_s_wait_asynccnt)
#define HAVE_WAIT_ASYNC_BUILTIN 1
#endif
#endif

#if defined(USE_ASYNC_STORE)
typedef int v4i __attribute__((ext_vector_type(4)));
typedef __attribute__((address_space(1))) v4i gv4i_t;   // global (device)
typedef __attribute__((address_space(3))) v4i lv4i_t;   // LDS (shared)
#endif

// ---------------------------------------------------------------------------
// Kernel 1: zero-fill the 304MB output volume.
// Async path: zeros live in a 4KB LDS block; each lane fires
// GLOBAL_STORE_ASYNC_FROM_LDS_B128 (ASYNCcnt-tracked, no VGPR data) in a
// grid-stride loop, then drains with s_wait_asynccnt 0.
// Fallback path: plain coalesced float4 (b128) stores.
// ---------------------------------------------------------------------------
__global__ __launch_bounds__(256) void zero_fill_kernel(float4* __restrict__ out,
                                                        long n4) {
    __shared__ float4 zbuf[256];
    zbuf[threadIdx.x] = make_float4(0.f, 0.f, 0.f, 0.f);
    __syncthreads();

    long stride = (long)gridDim.x * blockDim.x;
    long i = (long)blockIdx.x * blockDim.x + threadIdx.x;

#if defined(USE_ASYNC_STORE)
    lv4i_t* lsrc = (lv4i_t*)&zbuf[threadIdx.x];
    for (; i < n4; i += stride) {
        __builtin_amdgcn_global_store_async_from_lds_b128(
            (gv4i_t*)(out + i), lsrc, /*offset=*/0, /*cpol=*/0);
    }
#if defined(HAVE_WAIT_ASYNC_BUILTIN)
    __builtin_amdgcn_s_wait_asynccnt(0);
#else
    asm volatile("s_wait_asynccnt 0" ::: "memory");
#endif
#else
    const float4 z = make_float4(0.f, 0.f, 0.f, 0.f);
    for (; i < n4; i += stride) out[i] = z;
#endif
}

__global__ void zero_tail_kernel(float* __restrict__ out, long start, long n) {
    long i = start + blockIdx.x * blockDim.x + threadIdx.x;
    if (i < n) out[i] = 0.f;
}

// ---------------------------------------------------------------------------
// Kernel 2: Gaussian splat. One thread per (atom, window-cell).
//   val = exp(-((cx-x)^2 + (cy-y)^2 + (cz-z)^2))     (RES = 1.0)
//   out[b, t, cx, cy, cz] += val   (hardware fp32 global atomic)
// Type resolution mirrors jnp.searchsorted(off, a, 'right')-1 clipped to
// [0, NTYPES-1], then validity check a in [off[t], off[t]+num[t]).
// ---------------------------------------------------------------------------
__global__ void splat_kernel(const float* __restrict__ coords,
                             const int* __restrict__ nums,
                             const int* __restrict__ offs,
                             float* __restrict__ out,
                             int natoms, int total) {
    int id = blockIdx.x * blockDim.x + threadIdx.x;
    if (id >= total) return;

    int w    = id % WIN;
    int atom = id / WIN;
    int b    = atom / natoms;
    int a    = atom - b * natoms;

    const int* ob = offs + b * NTYPES;
    const int* nb = nums + b * NTYPES;
    int t = 0;
#pragma unroll
    for (int i = 1; i < NTYPES; ++i) t = (a >= ob[i]) ? i : t;
    bool valid = (a >= ob[t]) && (a < ob[t] + nb[t]);

    const float* c = coords + ((long)b * natoms + a) * 3;
    float x = c[0], y = c[1], z = c[2];
    int bx = (int)floorf(x);
    int by = (int)floorf(y);
    int bz = (int)floorf(z);

    // decompose w -> (dx,dy,dz) in [-2,2]^3 (ordering irrelevant for scatter)
    int qx = w / 25;
    int r  = w - qx * 25;
    int qy = r / 5;
    int qz = r - qy * 5;
    int cx = bx + qx - 2;
    int cy = by + qy - 2;
    int cz = bz + qz - 2;

    bool inb = (cx >= 0) & (cx < BOX) & (cy >= 0) & (cy < BOX) &
               (cz >= 0) & (cz < BOX);
    if (valid && inb) {
        float fx = (float)cx - x;
        float fy = (float)cy - y;
        float fz = (float)cz - z;
        float val = expf(-(fx * fx + fy * fy + fz * fz));
        long idx = ((((long)b * NTYPES + t) * BOX + cx) * BOX + cy) * BOX + cz;
        unsafeAtomicAdd(out + idx, val);   // -> global_atomic_add_f32
    }
}

// ---------------------------------------------------------------------------
// Launch: inputs are [coords f32 (B,N*3)], [num_atoms_of_type i32 (B,11)],
// [offsets i32 (B,11)]. Output f32 (B,11,120,120,120) flattened.
// ---------------------------------------------------------------------------
extern "C" void kernel_launch(void* const* d_in, const int* in_sizes, int n_in,
                              void* d_out, int out_size, void* d_ws, size_t ws_size,
                              hipStream_t stream) {
    const float* coords = (const float*)d_in[0];
    const int*   nums   = (const int*)d_in[1];
    const int*   offs   = (const int*)d_in[2];
    float*       out    = (float*)d_out;

    const int B      = in_sizes[1] / NTYPES;            // 4
    const int natoms = in_sizes[0] / (B * 3);           // 2200

    // 1) zero the volume (async LDS->global b128 stream, grid-stride)
    long n4 = (long)out_size / 4;
    int zblocks = 4096;                                  // ~1M threads, ~18 iters
    zero_fill_kernel<<<zblocks, 256, 0, stream>>>((float4*)out, n4);
    long tail_start = n4 * 4;
    long tail = (long)out_size - tail_start;
    if (tail > 0) {
        zero_tail_kernel<<<1, 256, 0, stream>>>(out, tail_start, (long)out_size);
    }

    // 2) splat
    int total = B * natoms * WIN;                        // 1,100,000
    int sblocks = (total + 255) / 256;
    splat_kernel<<<sblocks, 256, 0, stream>>>(coords, nums, offs, out,
                                              natoms, total);
}